// AdaptiveTopK_69312182223453
// MI455X (gfx1250) — compile-verified
//
#include <hip/hip_runtime.h>
#include <stdint.h>

// ---------------- problem constants (from the reference) ----------------
#define L_DIM   4096          // row length (and sequence length)
#define K_OUT   1024          // static top-k bound = min(K_MAX, L)
#define THREADS 256           // 8 wave32 waves per workgroup
#define NWAVES  (THREADS / 32)
#define KBASE   512.0f
#define KMINF   128.0f
#define BETAF   0.1f

// Pack a float score + index into a 64-bit key such that sorting keys in
// DESCENDING order reproduces lax.top_k ordering (descending value, ties ->
// smaller index first).  mono(f) is order-preserving over all finite floats.
__device__ __forceinline__ unsigned long long make_key(float f, int idx) {
    unsigned u    = __float_as_uint(f);
    unsigned flip = (unsigned)((int)u >> 31) | 0x80000000u; // neg: ~, pos: flip sign
    unsigned mono = u ^ flip;
    return ((unsigned long long)mono << 12) |
           (unsigned long long)(unsigned)(L_DIM - 1 - idx);
}

__global__ __launch_bounds__(THREADS)
void adaptive_topk_kernel(const float* __restrict__ scores,
                          int* __restrict__ out_idx,
                          int* __restrict__ out_k) {
    __shared__ float               srow[L_DIM];     // 16 KB staged row
    __shared__ unsigned long long  keys[L_DIM];     // 32 KB sort keys
    __shared__ float               red_s[NWAVES];   // cross-wave partials
    __shared__ float               red_q[NWAVES];

    const int row  = blockIdx.x;           // 0 .. B*L-1
    const int tid  = threadIdx.x;
    const int lane = tid & 31;             // wave32
    const int wave = tid >> 5;
    const float* gbase = scores + (size_t)row * L_DIM;

    // -------- Phase 1: async global -> LDS copy of the whole row ----------
    // CDNA5 path: GLOBAL_LOAD_ASYNC_TO_LDS_B128 (ASYNCcnt-tracked), no VGPR
    // round trip.  Read-once 256MB stream: TH_LOAD_NT so we don't thrash the
    // 192MB L2.  LDS byte address = low 32 bits of the flat shared pointer
    // (flat aperture mapping).
    {
        unsigned lds_base = (unsigned)(uintptr_t)(&srow[0]);
#pragma unroll
        for (int it = 0; it < L_DIM / (THREADS * 4); ++it) {   // 4 issues
            int elem = (it * THREADS + tid) * 4;               // vec4 of floats
            const float* ga = gbase + elem;
            unsigned     la = lds_base + (unsigned)elem * 4u;  // bytes
            asm volatile("global_load_async_to_lds_b128 %0, %1, off th:TH_LOAD_NT"
                         :: "v"(la), "v"(ga)
                         : "memory");
        }
        asm volatile("s_wait_asynccnt 0" ::: "memory");
    }
    __syncthreads();

    // -------- Phase 2: fused pass: variance partials + sort keys ----------
    {
        float s = 0.0f, q = 0.0f;
        for (int i = tid; i < L_DIM; i += THREADS) {
            float v = srow[i];
            s += v;
            q += v * v;
            keys[i] = make_key(v, i);
        }
        // wave32 butterfly reduction (deterministic), then one LDS combine
#pragma unroll
        for (int off = 16; off > 0; off >>= 1) {
            s += __shfl_down(s, off, 32);
            q += __shfl_down(q, off, 32);
        }
        if (lane == 0) {
            red_s[wave] = s;
            red_q[wave] = q;
        }
    }
    __syncthreads();

    if (tid == 0) {
        float sum = 0.0f, sumq = 0.0f;
#pragma unroll
        for (int w = 0; w < NWAVES; ++w) {
            sum  += red_s[w];
            sumq += red_q[w];
        }
        const float n  = (float)L_DIM;
        float var = (sumq - sum * sum / n) / (n - 1.0f);   // ddof = 1
        // softplus(x) = log(1 + e^x), numerically safe for large x
        float sp  = (var > 20.0f) ? var : logf(1.0f + expf(var));
        float kf  = KBASE * (1.0f + BETAF * sp);
        float kc  = fminf((float)K_OUT, fmaxf(KMINF, kf));
        __builtin_nontemporal_store((int)kc, &out_k[row]);  // truncation, like .int()
    }
    // keys[] fully written by Phase 2 loop; barrier before sorting
    __syncthreads();

    // -------- Phase 3: in-LDS bitonic sort, descending --------------------
    // 78 compare stages of 2048 pairs; 8 pairs/thread/stage.
    for (int k = 2; k <= L_DIM; k <<= 1) {
        for (int j = k >> 1; j > 0; j >>= 1) {
            for (int t = tid; t < (L_DIM / 2); t += THREADS) {
                int low = t & (j - 1);
                int i   = ((t ^ low) << 1) | low;   // first element of pair
                int p   = i | j;                    // partner
                unsigned long long a = keys[i];
                unsigned long long b = keys[p];
                // descending overall: blocks with (i & k)==0 sort descending
                bool swap = ((i & k) == 0) ? (a < b) : (a > b);
                if (swap) { keys[i] = b; keys[p] = a; }
            }
            __syncthreads();
        }
    }

    // -------- Phase 4: emit top-K indices (write-once -> NT stores) -------
    int* orow = out_idx + (size_t)row * K_OUT;
    for (int t = tid; t < K_OUT; t += THREADS) {
        unsigned packed = (unsigned)(keys[t] & 0xFFFull);
        __builtin_nontemporal_store(L_DIM - 1 - (int)packed, &orow[t]);
    }
}

extern "C" void kernel_launch(void* const* d_in, const int* in_sizes, int n_in,
                              void* d_out, int out_size, void* d_ws, size_t ws_size,
                              hipStream_t stream) {
    (void)n_in; (void)d_ws; (void)ws_size; (void)out_size;
    const float* scores = (const float*)d_in[0];

    // in_sizes[0] = B * L * L, with L fixed at 4096 by the reference setup
    long long total = (long long)in_sizes[0];
    int B    = (int)(total / ((long long)L_DIM * (long long)L_DIM));
    int rows = B * L_DIM;

    // d_out = [B, L, K_OUT] int32 indices, then [B, L] int32 k-values
    int* out_idx = (int*)d_out;
    int* out_k   = out_idx + (size_t)rows * K_OUT;

    adaptive_topk_kernel<<<dim3(rows), dim3(THREADS), 0, stream>>>(
        scores, out_idx, out_k);
}